// SkipGAT_71219147702914
// MI455X (gfx1250) — compile-verified
//
#include <hip/hip_runtime.h>
#include <math.h>

// ---------------------------------------------------------------------------
// SkipGAT for MI455X (gfx1250).
//   - Dense projections use native fp32 WMMA: v_wmma_f32_16x16x4_f32
//     (exact fp32 math, 16 WMMA steps per 16x16 tile, K=64).
//   - Edge softmax/aggregation: 3 passes with f32 global atomics; node feature
//     tables (~26 MB) are L2-resident (192 MB L2), so random gathers are cheap.
// ---------------------------------------------------------------------------

typedef __attribute__((ext_vector_type(2))) float v2f;
typedef __attribute__((ext_vector_type(8))) float v8f;

#define HC    64      // HEADS * OUT_CH
#define HEADS 4
#define NEG_ATT 0.2f
#define NEG_OUT 0.01f

// ---------------------------------------------------------------------------
// GEMM: Y[N,64] = X[N,64] @ W[64,64] + bias[64]
// One block = 128 threads = 4 waves; block handles a 16-row strip, wave w
// handles output columns [16w, 16w+16). Per ISA 7.12.2 fp32 layouts:
//   A (16x4):  lane<16 -> row=lane, {v0,v1}=K{k0,k0+1}; lane>=16 -> K{k0+2,k0+3}
//   B (4x16):  lane<16 -> col=lane, {v0,v1}=rows K{k0,k0+1}; lane>=16 -> K{k0+2,k0+3}
//   C (16x16): vgpr i, lane<16 -> (row i, col lane); lane>=16 -> (row i+8, col lane-16)
// Grid is sized to full tiles only; EXEC is all-ones for every WMMA.
// ---------------------------------------------------------------------------
__global__ __launch_bounds__(128) void gemm64_wmma(
    const float* __restrict__ X, const float* __restrict__ W,
    const float* __restrict__ bias, float* __restrict__ Y) {
  const int lane = threadIdx.x & 31;
  const int wave = threadIdx.x >> 5;          // 0..3
  const int rowBase = blockIdx.x * 16;
  const int colBase = wave * 16;
  const int half = lane >> 4;                 // 0 or 1
  const int l15  = lane & 15;

  v8f acc = {};
  const float* __restrict__ xrow = X + (size_t)(rowBase + l15) * HC;

#pragma unroll
  for (int k0 = 0; k0 < HC; k0 += 4) {
    const int ka = k0 + half * 2;
    v2f a, b;
    a.x = xrow[ka];
    a.y = xrow[ka + 1];
    b.x = W[(size_t)ka * HC + colBase + l15];
    b.y = W[(size_t)(ka + 1) * HC + colBase + l15];
    acc = __builtin_amdgcn_wmma_f32_16x16x4_f32(
        /*neg_a=*/false, a, /*neg_b=*/false, b,
        /*c_mod=*/(short)0, acc, /*reuse_a=*/false, /*reuse_b=*/false);
  }

  const int col = colBase + l15;
  const float bv = bias[col];
#pragma unroll
  for (int i = 0; i < 8; ++i) {
    const int row = rowBase + i + half * 8;
    Y[(size_t)row * HC + col] = acc[i] + bv;
  }
}

// Scalar tail for rows not covered by full 16-row tiles (not used for the
// reference sizes, which are multiples of 16, but kept for generality).
__global__ void gemm64_tail(const float* __restrict__ X, const float* __restrict__ W,
                            const float* __restrict__ bias, float* __restrict__ Y,
                            int rowStart, int nRows) {
  int idx = blockIdx.x * blockDim.x + threadIdx.x;
  if (idx >= nRows * HC) return;
  int r = rowStart + idx / HC, c = idx % HC;
  float s = bias[c];
  for (int k = 0; k < HC; ++k) s += X[(size_t)r * HC + k] * W[(size_t)k * HC + c];
  Y[(size_t)r * HC + c] = s;
}

// ---------------------------------------------------------------------------
// Edge-phase helpers
// ---------------------------------------------------------------------------
__device__ __forceinline__ void compute_logits(const float* __restrict__ xl_s,
                                               const float* __restrict__ xr_d,
                                               const float* __restrict__ att,
                                               float* __restrict__ lg) {
#pragma unroll
  for (int h = 0; h < HEADS; ++h) {
    float acc = 0.f;
#pragma unroll
    for (int q = 0; q < 4; ++q) {
      const float4 a = reinterpret_cast<const float4*>(xl_s)[h * 4 + q];
      const float4 b = reinterpret_cast<const float4*>(xr_d)[h * 4 + q];
      const float4 w = reinterpret_cast<const float4*>(att)[h * 4 + q];
      float v0 = a.x + b.x; v0 = v0 > 0.f ? v0 : NEG_ATT * v0;
      float v1 = a.y + b.y; v1 = v1 > 0.f ? v1 : NEG_ATT * v1;
      float v2 = a.z + b.z; v2 = v2 > 0.f ? v2 : NEG_ATT * v2;
      float v3 = a.w + b.w; v3 = v3 > 0.f ? v3 : NEG_ATT * v3;
      acc += v0 * w.x + v1 * w.y + v2 * w.z + v3 * w.w;
    }
    lg[h] = acc;
  }
}

__device__ __forceinline__ void atomicMaxF32(float* addr, float v) {
  // Monotonic int/uint mapping of IEEE f32 (init value is -inf).
  if (v >= 0.f) atomicMax((int*)addr, __float_as_int(v));
  else          atomicMin((unsigned int*)addr, __float_as_uint(v));
}

__device__ __forceinline__ bool edge_ids(int e, int E, int nSelf,
                                         const int* __restrict__ src,
                                         const int* __restrict__ dst,
                                         int& s, int& d) {
  if (e >= E + nSelf) return false;
  if (e < E) { s = src[e]; d = dst[e]; }
  else       { s = e - E;  d = s; }       // self loops
  return true;
}

// Pass 1: segment max of logits over dst
__global__ void edge_max_kernel(const float* __restrict__ xl, const float* __restrict__ xr,
                                const int* __restrict__ src, const int* __restrict__ dst,
                                const float* __restrict__ att, float* __restrict__ m,
                                int E, int nSelf) {
  int e = blockIdx.x * blockDim.x + threadIdx.x, s, d;
  if (!edge_ids(e, E, nSelf, src, dst, s, d)) return;
  float lg[HEADS];
  compute_logits(xl + (size_t)s * HC, xr + (size_t)d * HC, att, lg);
#pragma unroll
  for (int h = 0; h < HEADS; ++h) atomicMaxF32(&m[(size_t)d * HEADS + h], lg[h]);
}

// Pass 2: denom = segment sum of exp(logit - m[dst])
__global__ void edge_denom_kernel(const float* __restrict__ xl, const float* __restrict__ xr,
                                  const int* __restrict__ src, const int* __restrict__ dst,
                                  const float* __restrict__ att, const float* __restrict__ m,
                                  float* __restrict__ denom, int E, int nSelf) {
  int e = blockIdx.x * blockDim.x + threadIdx.x, s, d;
  if (!edge_ids(e, E, nSelf, src, dst, s, d)) return;
  float lg[HEADS];
  compute_logits(xl + (size_t)s * HC, xr + (size_t)d * HC, att, lg);
#pragma unroll
  for (int h = 0; h < HEADS; ++h)
    atomicAdd(&denom[(size_t)d * HEADS + h], expf(lg[h] - m[(size_t)d * HEADS + h]));
}

// Pass 3: out[dst] += alpha * xl[src]   (alpha = softmax weight per head)
__global__ void edge_aggr_kernel(const float* __restrict__ xl, const float* __restrict__ xr,
                                 const int* __restrict__ src, const int* __restrict__ dst,
                                 const float* __restrict__ att, const float* __restrict__ m,
                                 const float* __restrict__ denom, float* __restrict__ out,
                                 int E, int nSelf) {
  int e = blockIdx.x * blockDim.x + threadIdx.x, s, d;
  if (!edge_ids(e, E, nSelf, src, dst, s, d)) return;
  float lg[HEADS];
  compute_logits(xl + (size_t)s * HC, xr + (size_t)d * HC, att, lg);
  float alpha[HEADS];
#pragma unroll
  for (int h = 0; h < HEADS; ++h) {
    float u = expf(lg[h] - m[(size_t)d * HEADS + h]);
    alpha[h] = u / fmaxf(denom[(size_t)d * HEADS + h], 1e-16f);
  }
  const float* __restrict__ xs = xl + (size_t)s * HC;
  float* __restrict__ od = out + (size_t)d * HC;
#pragma unroll
  for (int c = 0; c < HC; ++c) atomicAdd(&od[c], alpha[c >> 4] * xs[c]);
}

__global__ void fill_kernel(float* __restrict__ p, float v, int n) {
  int i = blockIdx.x * blockDim.x + threadIdx.x;
  if (i < n) p[i] = v;
}

// Final: out = leaky_relu(out + conv_bias, 0.01)
__global__ void bias_lrelu_kernel(float* __restrict__ out, const float* __restrict__ bias, int n) {
  int i = blockIdx.x * blockDim.x + threadIdx.x;
  if (i >= n) return;
  float v = out[i] + bias[i & (HC - 1)];
  out[i] = v > 0.f ? v : NEG_OUT * v;
}

// ---------------------------------------------------------------------------
static inline void launch_gemm(const float* X, const float* W, const float* b,
                               float* Y, int N, hipStream_t stream) {
  int tiles = N / 16;
  if (tiles > 0) gemm64_wmma<<<tiles, 128, 0, stream>>>(X, W, b, Y);
  int rem = N - tiles * 16;
  if (rem > 0)
    gemm64_tail<<<(rem * HC + 255) / 256, 256, 0, stream>>>(X, W, b, Y, tiles * 16, rem);
}

extern "C" void kernel_launch(void* const* d_in, const int* in_sizes, int n_in,
                              void* d_out, int out_size, void* d_ws, size_t ws_size,
                              hipStream_t stream) {
  const float* x_tx    = (const float*)d_in[0];
  const float* x_bd    = (const float*)d_in[1];
  const int*   nn_src  = (const int*)d_in[2];
  const int*   nn_dst  = (const int*)d_in[3];
  const int*   tb_src  = (const int*)d_in[4];
  const int*   tb_dst  = (const int*)d_in[5];
  const float* Wl_nn   = (const float*)d_in[6];
  const float* bl_nn   = (const float*)d_in[7];
  const float* Wr_nn   = (const float*)d_in[8];
  const float* br_nn   = (const float*)d_in[9];
  const float* att_nn  = (const float*)d_in[10];
  const float* bias_nn = (const float*)d_in[11];
  const float* Wl_tb   = (const float*)d_in[12];
  const float* bl_tb   = (const float*)d_in[13];
  const float* Wr_tb   = (const float*)d_in[14];
  const float* br_tb   = (const float*)d_in[15];
  const float* att_tb  = (const float*)d_in[16];
  const float* bias_tb = (const float*)d_in[17];
  const float* Wlin_tx = (const float*)d_in[18];
  const float* blin_tx = (const float*)d_in[19];
  const float* Wlin_bd = (const float*)d_in[20];
  const float* blin_bd = (const float*)d_in[21];

  const int n_tx = in_sizes[0] / HC;   // IN_CH == 64 == HC
  const int n_bd = in_sizes[1] / HC;
  const int E_nn = in_sizes[2];
  const int E_tb = in_sizes[4];

  // Workspace layout (floats): ~83 MB total for reference sizes.
  float* ws = (float*)d_ws;
  float* xl_nn = ws;                         ws += (size_t)n_tx * HC;
  float* xr_nn = ws;                         ws += (size_t)n_tx * HC;
  float* xl_tb = ws;                         ws += (size_t)n_tx * HC;
  float* xr_tb = ws;                         ws += (size_t)n_bd * HC;
  float* m_nn  = ws;                         ws += (size_t)n_tx * HEADS;
  float* dn_nn = ws;                         ws += (size_t)n_tx * HEADS;
  float* m_tb  = ws;                         ws += (size_t)n_bd * HEADS;
  float* dn_tb = ws;                         ws += (size_t)n_bd * HEADS;

  float* out_tx = (float*)d_out;
  float* out_bd = out_tx + (size_t)n_tx * HC;

  // 1) Dense projections (WMMA fp32). Skip-linear writes straight into d_out.
  launch_gemm(x_tx, Wl_nn,   bl_nn,   xl_nn,  n_tx, stream);
  launch_gemm(x_tx, Wr_nn,   br_nn,   xr_nn,  n_tx, stream);
  launch_gemm(x_tx, Wl_tb,   bl_tb,   xl_tb,  n_tx, stream);
  launch_gemm(x_bd, Wr_tb,   br_tb,   xr_tb,  n_bd, stream);
  launch_gemm(x_tx, Wlin_tx, blin_tx, out_tx, n_tx, stream);
  launch_gemm(x_bd, Wlin_bd, blin_bd, out_bd, n_bd, stream);

  // 2) Init segment-max (-inf) and denominators (0).
  fill_kernel<<<(n_tx * HEADS + 255) / 256, 256, 0, stream>>>(m_nn, -INFINITY, n_tx * HEADS);
  fill_kernel<<<(n_tx * HEADS + 255) / 256, 256, 0, stream>>>(dn_nn, 0.f, n_tx * HEADS);
  fill_kernel<<<(n_bd * HEADS + 255) / 256, 256, 0, stream>>>(m_tb, -INFINITY, n_bd * HEADS);
  fill_kernel<<<(n_bd * HEADS + 255) / 256, 256, 0, stream>>>(dn_tb, 0.f, n_bd * HEADS);

  // 3) tx->tx relation (with self loops appended, matching add_self_loops=True).
  {
    int tot = E_nn + n_tx, blk = (tot + 255) / 256;
    edge_max_kernel  <<<blk, 256, 0, stream>>>(xl_nn, xr_nn, nn_src, nn_dst, att_nn, m_nn, E_nn, n_tx);
    edge_denom_kernel<<<blk, 256, 0, stream>>>(xl_nn, xr_nn, nn_src, nn_dst, att_nn, m_nn, dn_nn, E_nn, n_tx);
    edge_aggr_kernel <<<blk, 256, 0, stream>>>(xl_nn, xr_nn, nn_src, nn_dst, att_nn, m_nn, dn_nn, out_tx, E_nn, n_tx);
  }
  // 4) tx->bd relation (no self loops).
  {
    int blk = (E_tb + 255) / 256;
    edge_max_kernel  <<<blk, 256, 0, stream>>>(xl_tb, xr_tb, tb_src, tb_dst, att_tb, m_tb, E_tb, 0);
    edge_denom_kernel<<<blk, 256, 0, stream>>>(xl_tb, xr_tb, tb_src, tb_dst, att_tb, m_tb, dn_tb, E_tb, 0);
    edge_aggr_kernel <<<blk, 256, 0, stream>>>(xl_tb, xr_tb, tb_src, tb_dst, att_tb, m_tb, dn_tb, out_bd, E_tb, 0);
  }

  // 5) Final skip-add bias + leaky_relu(0.01), in place on d_out.
  bias_lrelu_kernel<<<((size_t)n_tx * HC + 255) / 256, 256, 0, stream>>>(out_tx, bias_nn, n_tx * HC);
  bias_lrelu_kernel<<<((size_t)n_bd * HC + 255) / 256, 256, 0, stream>>>(out_bd, bias_tb, n_bd * HC);
}